// Proto_Metric_Semi_31937376813212
// MI455X (gfx1250) — compile-verified
//
#include <hip/hip_runtime.h>

typedef __attribute__((ext_vector_type(2))) float v2f;
typedef __attribute__((ext_vector_type(8))) float v8f;

#define B_Q    4096
#define D_DIM  1600
#define NCLS   64
#define SHOT_N 16
#define U_DIM  8192

// ---------------- kernel 1: denom[c] = SHOT + sum_u w[u][c] ----------------
__global__ void __launch_bounds__(256) k_denom(const float* __restrict__ w,
                                               float* __restrict__ denom) {
  __shared__ float red[256];
  const int c = blockIdx.x;
  float s = 0.f;
  for (int u = threadIdx.x; u < U_DIM; u += 256) s += w[(size_t)u * NCLS + c];
  red[threadIdx.x] = s;
  __syncthreads();
  for (int off = 128; off > 0; off >>= 1) {
    if (threadIdx.x < off) red[threadIdx.x] += red[threadIdx.x + off];
    __syncthreads();
  }
  if (threadIdx.x == 0) denom[c] = (float)SHOT_N + red[0];
}

// ------ kernel 2: proto via WMMA GEMM (M=64 classes, N=1600, K=8192) -------
// Pair-interleaved LDS: element (k,col) -> lds[(k>>1)*160 + 2*col + (k&1)],
// so each WMMA fragment {K,K+1} is one aligned ds_load_b64 into an adjacent
// VGPR pair (no repacking movs). Pitch 160 == 32 mod 64 banks, so the two
// lane-halves (pair p vs p+1) hit disjoint 32-bank halves.
__global__ void __launch_bounds__(128)
k_proto(const float* __restrict__ w, const float* __restrict__ x3,
        const float* __restrict__ x2, const float* __restrict__ denom,
        float* __restrict__ proto) {
  __shared__ float ldsW[32 * 160];  // 32 k-pairs x 64 m interleaved
  __shared__ float ldsB[32 * 160];  // 32 k-pairs x 64 n interleaved
  const int tid  = threadIdx.x;
  const int wave = tid >> 5;  // wave32
  const int lane = tid & 31;
  const int half = lane >> 4;
  const int l    = lane & 15;
  const int n0   = blockIdx.x * 64;  // 25 blocks cover D=1600
  const int m0   = wave * 16;        // 4 waves cover C=64

  v8f acc[4] = {{}, {}, {}, {}};
  for (int k0 = 0; k0 < U_DIM; k0 += 64) {
    // stage + interleave w[k0..k0+63][0..63]: 512 pair-tasks, 4 per thread
    for (int i = tid; i < 512; i += 128) {
      const int p = i >> 4, c4 = i & 15;
      const float4 e = *(const float4*)&w[(size_t)(k0 + 2 * p) * NCLS + c4 * 4];
      const float4 o =
          *(const float4*)&w[(size_t)(k0 + 2 * p + 1) * NCLS + c4 * 4];
      *(float4*)&ldsW[p * 160 + c4 * 8 + 0] = make_float4(e.x, o.x, e.y, o.y);
      *(float4*)&ldsW[p * 160 + c4 * 8 + 4] = make_float4(e.z, o.z, e.w, o.w);
    }
    // stage + interleave x3[k0..k0+63][n0..n0+63]
    for (int i = tid; i < 512; i += 128) {
      const int p = i >> 4, c4 = i & 15;
      const float4 e =
          *(const float4*)&x3[(size_t)(k0 + 2 * p) * D_DIM + n0 + c4 * 4];
      const float4 o =
          *(const float4*)&x3[(size_t)(k0 + 2 * p + 1) * D_DIM + n0 + c4 * 4];
      *(float4*)&ldsB[p * 160 + c4 * 8 + 0] = make_float4(e.x, o.x, e.y, o.y);
      *(float4*)&ldsB[p * 160 + c4 * 8 + 4] = make_float4(e.z, o.z, e.w, o.w);
    }
    if (k0 + 64 < U_DIM) {
      __builtin_prefetch(&w[(size_t)(k0 + 64 + (tid >> 1)) * NCLS], 0, 1);
      __builtin_prefetch(&x3[(size_t)(k0 + 64 + tid) * D_DIM + n0], 0, 1);
    }
    __syncthreads();
#pragma unroll
    for (int kk = 0; kk < 64; kk += 4) {
      const int p = (kk >> 1) + half;  // pair index for K = kk + 2*half
      const v2f a = *(const v2f*)&ldsW[p * 160 + (m0 + l) * 2];  // ds_load_b64
#pragma unroll
      for (int nt = 0; nt < 4; ++nt) {
        const v2f b = *(const v2f*)&ldsB[p * 160 + (nt * 16 + l) * 2];
        acc[nt] = __builtin_amdgcn_wmma_f32_16x16x4_f32(
            false, a, false, b, (short)0, acc[nt], false, false);
      }
    }
    __syncthreads();
  }

  // epilogue: proto = (support_sum + acc) / denom
#pragma unroll
  for (int nt = 0; nt < 4; ++nt) {
    const int n = n0 + nt * 16 + l;
#pragma unroll
    for (int r = 0; r < 8; ++r) {
      const int m = m0 + r + half * 8;  // C/D layout: VGPR r holds M=r, M=r+8
      float s = 0.f;
#pragma unroll
      for (int sh = 0; sh < SHOT_N; ++sh)
        s += x2[((size_t)m * SHOT_N + sh) * D_DIM + n];
      proto[(size_t)m * D_DIM + n] = (acc[nt][r] + s) / denom[m];
    }
  }
}

// ---------------- kernel 3: out[row] = sum_col x[row][col]^2 ----------------
__global__ void __launch_bounds__(256) k_rowsq(const float* __restrict__ x,
                                               int ncols,
                                               float* __restrict__ out) {
  __shared__ float red[256];
  const int row = blockIdx.x;
  float s = 0.f;
  for (int c = threadIdx.x; c < ncols; c += 256) {
    const float v = x[(size_t)row * ncols + c];
    s += v * v;
  }
  red[threadIdx.x] = s;
  __syncthreads();
  for (int off = 128; off > 0; off >>= 1) {
    if (threadIdx.x < off) red[threadIdx.x] += red[threadIdx.x + off];
    __syncthreads();
  }
  if (threadIdx.x == 0) out[row] = red[0];
}

// ---- kernel 4: score via WMMA GEMM (M=4096, N=64, K=1600), fused epilogue --
// [row][k] LDS layout (stride 68): fragments are aligned contiguous K-pairs
// -> single ds_load_b64 each, bank-conflict-free (banks 4*row + k; the two
// lane-halves differ by 2 mod 4). Each wave: one m-tile x all 4 n-tiles,
// 4 independent accumulators, A fragment reused 4x.
__global__ void __launch_bounds__(128)
k_score(const float* __restrict__ q, const float* __restrict__ proto,
        const float* __restrict__ q2, const float* __restrict__ p2,
        float* __restrict__ out) {
  __shared__ float ldsQ[64 * 68];  // [m][k]
  __shared__ float ldsP[64 * 68];  // [n][k]
  const int tid  = threadIdx.x;
  const int wave = tid >> 5;
  const int lane = tid & 31;
  const int half = lane >> 4;
  const int l    = lane & 15;
  const int m0   = blockIdx.x * 64;  // 64 blocks cover B=4096
  const int lm   = wave * 16 + l;    // wave's local m row

  v8f acc[4] = {{}, {}, {}, {}};
  for (int k0 = 0; k0 < D_DIM; k0 += 64) {  // 25 chunks
    // stage q[m0..m0+63][k0..k0+63]  (1024 float4, 8 per thread)
    for (int i = tid; i < 64 * 16; i += 128) {
      const int r = i >> 4, c4 = i & 15;
      *(float4*)&ldsQ[r * 68 + c4 * 4] =
          *(const float4*)&q[(size_t)(m0 + r) * D_DIM + k0 + c4 * 4];
    }
    // stage proto[0..63][k0..k0+63]  (1024 float4, 8 per thread)
    for (int i = tid; i < 64 * 16; i += 128) {
      const int r = i >> 4, c4 = i & 15;
      *(float4*)&ldsP[r * 68 + c4 * 4] =
          *(const float4*)&proto[(size_t)r * D_DIM + k0 + c4 * 4];
    }
    if (k0 + 64 < D_DIM) {
      __builtin_prefetch(&q[(size_t)(m0 + (tid >> 1)) * D_DIM + k0 + 64], 0, 1);
      __builtin_prefetch(&proto[(size_t)(tid >> 1) * D_DIM + k0 + 64], 0, 1);
    }
    __syncthreads();
#pragma unroll
    for (int kk = 0; kk < 64; kk += 4) {
      const int kr = kk + 2 * half;
      const v2f a = *(const v2f*)&ldsQ[lm * 68 + kr];  // ds_load_b64
#pragma unroll
      for (int nt = 0; nt < 4; ++nt) {
        const v2f b = *(const v2f*)&ldsP[(nt * 16 + l) * 68 + kr];
        acc[nt] = __builtin_amdgcn_wmma_f32_16x16x4_f32(
            false, a, false, b, (short)0, acc[nt], false, false);
      }
    }
    __syncthreads();
  }

  // epilogue: out[m][n] = -(q2[m] + p2[n] - 2*cross)
#pragma unroll
  for (int nt = 0; nt < 4; ++nt) {
    const int n   = nt * 16 + l;
    const float pn = p2[n];
#pragma unroll
    for (int r = 0; r < 8; ++r) {
      const int m = m0 + wave * 16 + r + half * 8;
      out[(size_t)m * NCLS + n] = -(q2[m] + pn - 2.0f * acc[nt][r]);
    }
  }
}

extern "C" void kernel_launch(void* const* d_in, const int* in_sizes, int n_in,
                              void* d_out, int out_size, void* d_ws,
                              size_t ws_size, hipStream_t stream) {
  const float* x1 = (const float*)d_in[0];  // [B, D] queries
  const float* x2 = (const float*)d_in[1];  // [C, SHOT, D] support
  const float* x3 = (const float*)d_in[2];  // [U, D] unlabeled
  const float* w  = (const float*)d_in[3];  // [U, C]
  float* out = (float*)d_out;               // [B, C]

  float* ws    = (float*)d_ws;
  float* proto = ws;           // 64*1600 = 102400 floats
  float* denom = ws + 102400;  // 64
  float* p2    = ws + 102464;  // 64
  float* q2    = ws + 102528;  // 4096   (total ~417 KB)

  k_denom<<<NCLS, 256, 0, stream>>>(w, denom);
  k_proto<<<D_DIM / 64, 128, 0, stream>>>(w, x3, x2, denom, proto);
  k_rowsq<<<NCLS, 256, 0, stream>>>(proto, D_DIM, p2);
  k_rowsq<<<B_Q, 256, 0, stream>>>(x1, D_DIM, q2);
  k_score<<<B_Q / 64, 128, 0, stream>>>(x1, proto, q2, p2, out);
}